// WarpMeshRasterizer_6949257085071
// MI455X (gfx1250) — compile-verified
//
#include <hip/hip_runtime.h>

typedef float v2f __attribute__((ext_vector_type(2)));
typedef float v8f __attribute__((ext_vector_type(8)));

#define IMG   512
#define NTRI  2048
#define BIGF  1.0e10f

// ---------------------------------------------------------------------------
// Kernel 1: vertex transform  clip = [v,1] @ mvp^T  (+ ortho z/6 fix-up)
// ---------------------------------------------------------------------------
__global__ void vtx_xform_kernel(const float* __restrict__ vtx,
                                 const float* __restrict__ mvp,
                                 const int*   __restrict__ is_ortho,
                                 float4*      __restrict__ clip4,
                                 int nv) {
    int i = blockIdx.x * blockDim.x + threadIdx.x;
    if (i >= nv) return;
    float x = vtx[3 * i + 0], y = vtx[3 * i + 1], z = vtx[3 * i + 2];
    float cx = mvp[0]  * x + mvp[1]  * y + mvp[2]  * z + mvp[3];
    float cy = mvp[4]  * x + mvp[5]  * y + mvp[6]  * z + mvp[7];
    float cz = mvp[8]  * x + mvp[9]  * y + mvp[10] * z + mvp[11];
    float cw = mvp[12] * x + mvp[13] * y + mvp[14] * z + mvp[15];
    if (is_ortho[0]) cz *= (1.0f / 6.0f);
    clip4[i] = make_float4(cx, cy, cz, cw);
}

// ---------------------------------------------------------------------------
// Kernel 2: per-triangle screen-space affine coefficients.
//   W0 = e0x*px + e0y*py + c0  (1/denom folded in), same for W1;
//   z  = zx*px  + zy*py  + zc  (since z = W0*(z0-z2)+W1*(z1-z2)+z2 is affine).
// Degenerate triangles (|denom| < 1e-8) get c0=c1=-2e30 -> inside test fails.
// Also stores clip-space (w0,w1,w2) and (z0,z1,z2) for perspective depth.
// ---------------------------------------------------------------------------
__device__ __forceinline__ float3 ndc_of(float4 c) {
    if (c.w != 0.0f) {
        float inv = 1.0f / c.w;
        return make_float3(c.x * inv, c.y * inv, c.z * inv);
    }
    return make_float3(c.x, c.y, c.z);
}

__global__ void tri_setup_kernel(const int*    __restrict__ tidx,
                                 const float4* __restrict__ clip4,
                                 float4* __restrict__ W0c,
                                 float4* __restrict__ W1c,
                                 float4* __restrict__ Zc,
                                 float4* __restrict__ Wclip,
                                 float4* __restrict__ Zclip,
                                 int nt, int img) {
    int t = blockIdx.x * blockDim.x + threadIdx.x;
    if (t >= nt) return;
    int i0 = tidx[3 * t + 0], i1 = tidx[3 * t + 1], i2 = tidx[3 * t + 2];
    float4 cA = clip4[i0], cB = clip4[i1], cC = clip4[i2];
    float3 n0 = ndc_of(cA), n1 = ndc_of(cB), n2 = ndc_of(cC);
    float hw = 0.5f * (float)img;
    float x0 = truncf((n0.x + 1.0f) * hw), y0 = truncf((n0.y + 1.0f) * hw), z0 = n0.z;
    float x1 = truncf((n1.x + 1.0f) * hw), y1 = truncf((n1.y + 1.0f) * hw), z1 = n1.z;
    float x2 = truncf((n2.x + 1.0f) * hw), y2 = truncf((n2.y + 1.0f) * hw), z2 = n2.z;

    float denom = (y1 - y2) * (x0 - x2) + (x2 - x1) * (y0 - y2);
    bool  ok = fabsf(denom) >= 1e-8f;
    float d  = ok ? denom : 1.0f;
    float rd = 1.0f / d;

    float e0x = (y1 - y2) * rd, e0y = (x2 - x1) * rd;
    float c0  = -(e0x * x2 + e0y * y2);
    float e1x = (y2 - y0) * rd, e1y = (x0 - x2) * rd;
    float c1  = -(e1x * x2 + e1y * y2);
    float dz0 = z0 - z2, dz1 = z1 - z2;
    float zx = e0x * dz0 + e1x * dz1;
    float zy = e0y * dz0 + e1y * dz1;
    float zcc = c0 * dz0 + c1 * dz1 + z2;

    if (!ok) {   // poison: W0/W1 evaluate to -2e30 everywhere -> never inside
        W0c[t] = make_float4(0.f, 0.f, -2.0e30f, 0.f);
        W1c[t] = make_float4(0.f, 0.f, -2.0e30f, 0.f);
        Zc[t]  = make_float4(0.f, 0.f, 0.f, 0.f);
    } else {
        W0c[t] = make_float4(e0x, e0y, c0, 0.f);
        W1c[t] = make_float4(e1x, e1y, c1, 0.f);
        Zc[t]  = make_float4(zx, zy, zcc, 0.f);
    }
    Wclip[t] = make_float4(cA.w, cB.w, cC.w, 0.f);
    Zclip[t] = make_float4(cA.z, cB.z, cC.z, 0.f);
}

// ---------------------------------------------------------------------------
// Kernel 3: WMMA raster. One wave = one 16-pixel row segment.
//   A (16x4 f32): rows = 16 pixels, K = [px, py, 1, 0]
//   B (4x16 f32): cols = 16 triangles, K = [ex, ey, c, 0]
//   D (16x16 f32) = numerator matrix; 3 WMMAs/chunk give W0, W1, z.
// Fully branchless inner loop: min3 inside-test + cndmask running min.
// Strict '<' keeps the lowest triangle index within a lane slot; the final
// 4-stage shuffle reduce tie-breaks on index across slots -> exact reference
// tie-break semantics (global argmin, lowest index wins ties).
// ---------------------------------------------------------------------------
__global__ void __launch_bounds__(256)
raster_kernel(const float4* __restrict__ W0c,
              const float4* __restrict__ W1c,
              const float4* __restrict__ Zc,
              int*   __restrict__ tid_img,
              float* __restrict__ b0_img,
              float* __restrict__ b1_img,
              int img, int nt) {
    int gtid = blockIdx.x * blockDim.x + threadIdx.x;
    int wave = gtid >> 5;
    int lane = threadIdx.x & 31;
    int half = lane >> 4;          // 0: K=0,1 slice   1: K=2,3 slice
    int l15  = lane & 15;

    int segsPerRow = img >> 4;     // 32
    int py   = wave / segsPerRow;
    int xbas = (wave % segsPerRow) << 4;

    // A-matrix fragment (16x4 f32 layout: lanes0-15 hold K=0,1; lanes16-31 K=2,3)
    v2f a;
    a.x = half ? 1.0f : (float)(xbas + l15);
    a.y = half ? 0.0f : (float)py;

    float bestZ[8];
    int   bestI[8];
    float bB0[8], bB1[8];
#pragma unroll
    for (int r = 0; r < 8; ++r) {
        bestZ[r] = BIGF; bestI[r] = 0x7fffffff; bB0[r] = 0.0f; bB1[r] = 0.0f;
    }

    int nChunk = nt >> 4;          // 128 chunks of 16 triangles
    for (int c = 0; c < nChunk; ++c) {
        int tri = (c << 4) + l15;
        float4 q0 = W0c[tri];
        float4 q1 = W1c[tri];
        float4 qz = Zc[tri];
        // Speculative prefetch of next chunk (stays inside the workspace;
        // past-the-end addresses are silently dropped per ISA).
        __builtin_prefetch(&W0c[tri + 16], 0, 3);
        __builtin_prefetch(&W1c[tri + 16], 0, 3);
        __builtin_prefetch(&Zc[tri + 16], 0, 3);

        v2f b0v, b1v, bzv;
        b0v.x = half ? q0.z : q0.x;  b0v.y = half ? 0.0f : q0.y;
        b1v.x = half ? q1.z : q1.x;  b1v.y = half ? 0.0f : q1.y;
        bzv.x = half ? qz.z : qz.x;  bzv.y = half ? 0.0f : qz.y;

        v8f cz = {};
        v8f accW0 = __builtin_amdgcn_wmma_f32_16x16x4_f32(
            false, a, false, b0v, (short)0, cz, false, false);
        v8f accW1 = __builtin_amdgcn_wmma_f32_16x16x4_f32(
            false, a, false, b1v, (short)0, cz, false, false);
        v8f accZ  = __builtin_amdgcn_wmma_f32_16x16x4_f32(
            false, a, false, bzv, (short)0, cz, false, false);

#pragma unroll
        for (int r = 0; r < 8; ++r) {
            float w0 = accW0[r];
            float w1 = accW1[r];
            float w2 = 1.0f - w0 - w1;
            float mn = fminf(fminf(w0, w1), w2);       // v_min3 path
            float zs = (mn >= 0.0f) ? accZ[r] : BIGF;  // cndmask
            bool take = zs < bestZ[r];
            bestZ[r] = take ? zs  : bestZ[r];
            bestI[r] = take ? tri : bestI[r];
            bB0[r]   = take ? w0  : bB0[r];
            bB1[r]   = take ? w1  : bB1[r];
        }
    }

    // Cross-lane min over the 16 triangle slots (within each half-wave group),
    // tie-break on lowest triangle index. Branchless selects.
#pragma unroll
    for (int m = 1; m < 16; m <<= 1) {
#pragma unroll
        for (int r = 0; r < 8; ++r) {
            float oz = __shfl_xor(bestZ[r], m, 32);
            int   oi = __shfl_xor(bestI[r], m, 32);
            float o0 = __shfl_xor(bB0[r], m, 32);
            float o1 = __shfl_xor(bB1[r], m, 32);
            bool take = (oz < bestZ[r]) || (oz == bestZ[r] && oi < bestI[r]);
            bestZ[r] = take ? oz : bestZ[r];
            bestI[r] = take ? oi : bestI[r];
            bB0[r]   = take ? o0 : bB0[r];
            bB1[r]   = take ? o1 : bB1[r];
        }
    }

    if (l15 == 0) {
        // lane 0 owns pixels M=0..7 (VGPR r), lane 16 owns M=8..15.
#pragma unroll
        for (int r = 0; r < 8; ++r) {
            int pxi = xbas + r + (half << 3);
            int o   = py * img + pxi;
            bool hit = bestZ[r] < BIGF;
            tid_img[o] = hit ? bestI[r] : -1;
            b0_img[o]  = hit ? bB0[r] : 0.0f;
            b1_img[o]  = hit ? bB1[r] : 0.0f;
        }
    }
}

// ---------------------------------------------------------------------------
// Kernel 4: edge detect / AA / perspective-correct depth, write outputs.
// ---------------------------------------------------------------------------
__global__ void finalize_kernel(const int*    __restrict__ tid_img,
                                const float*  __restrict__ b0_img,
                                const float*  __restrict__ b1_img,
                                const float4* __restrict__ Wclip,
                                const float4* __restrict__ Zclip,
                                float* __restrict__ out,
                                int img, int nt) {
    int p = blockIdx.x * blockDim.x + threadIdx.x;
    int npix = img * img;
    if (p >= npix) return;
    int x = p % img, y = p / img;

    int t = tid_img[p];
    bool valid = t >= 0;

    int yu = max(y - 1, 0), yd = min(y + 1, img - 1);
    int xl = max(x - 1, 0), xr = min(x + 1, img - 1);
    bool edge = (tid_img[yu * img + x] != t) | (tid_img[yd * img + x] != t) |
                (tid_img[y * img + xl] != t) | (tid_img[y * img + xr] != t);

    float b0 = valid ? b0_img[p] : 0.0f;
    float b1 = valid ? b1_img[p] : 0.0f;
    float b2 = valid ? (1.0f - b0 - b1) : 0.0f;

    float mb = fminf(b0, fminf(b1, b2));
    float alpha_edge = (mb < 0.5f) ? fminf(fmaxf(mb * 2.0f, 0.0f), 1.0f) : 1.0f;
    float aa = valid ? (edge ? alpha_edge : 1.0f) : 0.0f;

    int tc = valid ? min(t, nt - 1) : 0;
    float4 wc = Wclip[tc];
    float4 zc = Zclip[tc];
    bool allw = (wc.x != 0.0f) && (wc.y != 0.0f) && (wc.z != 0.0f);
    float zs0 = (zc.x == 0.0f) ? 1.0f : zc.x;
    float zs1 = (zc.y == 0.0f) ? 1.0f : zc.y;
    float zs2 = (zc.z == 0.0f) ? 1.0f : zc.z;
    float inv = b0 / zs0 + b1 / zs1 + b2 / zs2;
    float inv_safe = (inv == 0.0f) ? 1.0f : inv;
    float dpix = allw ? (1.0f / inv_safe)
                      : (b0 * zc.x + b1 * zc.y + b2 * zc.z);
    float depth_final = valid ? dpix : BIGF;

    // Output layout: depth[N] | tid[N] (int32) | bary[N*3] | aa[N]
    out[p] = depth_final;
    reinterpret_cast<int*>(out + npix)[p] = t;
    float* bary = out + 2 * npix;
    bary[3 * p + 0] = b0;
    bary[3 * p + 1] = b1;
    bary[3 * p + 2] = b2;
    out[5 * npix + p] = aa;
}

// ---------------------------------------------------------------------------
extern "C" void kernel_launch(void* const* d_in, const int* in_sizes, int n_in,
                              void* d_out, int out_size, void* d_ws, size_t ws_size,
                              hipStream_t stream) {
    const float* vertices = (const float*)d_in[0];
    const int*   tri_idx  = (const int*)d_in[1];
    const float* mvp      = (const float*)d_in[2];
    const int*   is_ortho = (const int*)d_in[4];

    int nv = in_sizes[0] / 3;
    int nt = in_sizes[1] / 3;
    const int img = IMG;

    // workspace layout (16B aligned)
    char* ws = (char*)d_ws;
    size_t need = 196608 + 3 * (size_t)img * img * 4;
    if (ws_size < need) return;
    float4* clip4   = (float4*)(ws + 0);        // nv*16         (< 32 KB)
    float4* W0c     = (float4*)(ws + 32768);    // nt*16 = 32 KB
    float4* W1c     = (float4*)(ws + 65536);
    float4* Zc      = (float4*)(ws + 98304);
    float4* Wclip   = (float4*)(ws + 131072);
    float4* Zclip   = (float4*)(ws + 163840);
    int*    tid_img = (int*)  (ws + 196608);                       // 1 MB
    float*  b0_img  = (float*)(ws + 196608 + (size_t)img * img * 4);
    float*  b1_img  = (float*)(ws + 196608 + (size_t)img * img * 8);

    vtx_xform_kernel<<<(nv + 127) / 128, 128, 0, stream>>>(
        vertices, mvp, is_ortho, clip4, nv);

    tri_setup_kernel<<<(nt + 127) / 128, 128, 0, stream>>>(
        tri_idx, clip4, W0c, W1c, Zc, Wclip, Zclip, nt, img);

    int waves   = (img * img) / 16;        // 16384 waves, one per 16-pixel row seg
    int threads = waves * 32;
    raster_kernel<<<threads / 256, 256, 0, stream>>>(
        W0c, W1c, Zc, tid_img, b0_img, b1_img, img, nt);

    finalize_kernel<<<(img * img + 255) / 256, 256, 0, stream>>>(
        tid_img, b0_img, b1_img, Wclip, Zclip, (float*)d_out, img, nt);
}